// WindAwareDoubleShifter_88519275970885
// MI455X (gfx1250) — compile-verified
//
#include <hip/hip_runtime.h>

// Problem constants (from reference setup_inputs)
#define B      2
#define PL     14
#define LAT    128
#define LON    256
#define CH     96
#define C4     (CH / 4)        // 24 float4 chunks per row
#define NDIRS  9
#define LATR   4
#define LONR   8
#define NREG   (B * LATR * LONR)   // 64 regions, 32x32 each

typedef float v4f __attribute__((ext_vector_type(4)));

// d_ws layout (ints):
//   [0 .. 63]          packed per-region shift ((latAdd<<8)|lonAdd)
//   [64 .. 64+576)     per-region 9-bin histograms (region-major)

// One block per region (64 blocks). Histogram 32x32 ids in LDS, argmax (first-max),
// store packed (latAdd<<8 | lonAdd) where latAdd = 3-2*lat_s, lonAdd = 6-2*lon_s.
__global__ __launch_bounds__(256) void wads_region_hist(const int* __restrict__ wid,
                                                        int* __restrict__ ws) {
    __shared__ int hist[NDIRS];
    const int t = threadIdx.x;
    if (t < NDIRS) hist[t] = 0;
    __syncthreads();

    const int blk  = blockIdx.x;         // b*32 + latR*8 + lonR
    const int lonR = blk & 7;
    const int latR = (blk >> 3) & 3;
    const int b    = blk >> 5;
    const int base = (b * LAT + latR * 32) * LON + lonR * 32;

    __builtin_prefetch(wid + base, 0, 0);    // gfx1250 global_prefetch_b8

#pragma unroll
    for (int k = 0; k < 4; ++k) {
        const int local = t + 256 * k;           // 0..1023 over the 32x32 region
        const int llat  = local >> 5;
        const int llon  = local & 31;
        const int d     = wid[base + llat * LON + llon];
        atomicAdd(&hist[d], 1);                  // LDS atomics only
    }
    __syncthreads();

    if (t < NDIRS) ws[64 + blk * NDIRS + t] = hist[t];   // publish regional bincount
    if (t == 0) {
        const int lat_s[NDIRS] = {0, -1, -1, 0, 1, 1, 1, 0, -1};
        const int lon_s[NDIRS] = {0,  0,  1, 1, 1, 0, -1, -1, -1};
        int best = 0, bestc = hist[0];
#pragma unroll
        for (int d = 1; d < NDIRS; ++d)
            if (hist[d] > bestc) { bestc = hist[d]; best = d; }   // strict > == first max
        const int latAdd = 3 - 2 * lat_s[best];   // in [1,5]
        const int lonAdd = 6 - 2 * lon_s[best];   // in [4,8]
        ws[blk] = (latAdd << 8) | lonAdd;
    }
}

// Hot kernel: one float4 per thread, NT streaming load/store (working set >> 192MB L2).
__global__ __launch_bounds__(256) void wads_gather(const float* __restrict__ x,
                                                   const int* __restrict__ rd,
                                                   float* __restrict__ out) {
    const int idx = blockIdx.x * 256 + threadIdx.x;   // grid is exact, no tail check

    const int c4  = idx % C4;
    int r         = idx / C4;            // flat row id (b,pl,lat,lon)
    const int lon = r & (LON - 1);  r >>= 8;
    const int lat = r & (LAT - 1);  r >>= 7;
    const int pl  = r % PL;
    const int b   = r / PL;

    // packed region shift: rd[b*32 + (lat/32)*8 + (lon/32)]  (256B table, L0-resident)
    const int p = rd[(b << 5) + ((lat >> 5) << 3) + (lon >> 5)];
    const int lat_src = (lat + (p >> 8))  & (LAT - 1);
    const int lon_src = (lon + (p & 255)) & (LON - 1);
    int pl_src = pl + 1; if (pl_src == PL) pl_src = 0;

    const int src = (((b * PL + pl_src) * LAT + lat_src) * LON + lon_src) * CH + c4 * 4;
    const int dst = idx * 4;             // identity layout on the destination side

    v4f v = __builtin_nontemporal_load((const v4f*)(x + src));
    __builtin_nontemporal_store(v, (v4f*)(out + dst));
}

// One wave: lanes 0..8 reduce the 64 regional histograms, lane 0 takes first-max.
__global__ __launch_bounds__(32) void wads_finalize(const int* __restrict__ ws,
                                                    float* __restrict__ out_last) {
    __shared__ int sums[NDIRS];
    const int t = threadIdx.x;
    if (t < NDIRS) {
        int s = 0;
#pragma unroll
        for (int r = 0; r < NREG; ++r) s += ws[64 + r * NDIRS + t];
        sums[t] = s;
    }
    __syncthreads();
    if (t == 0) {
        int best = 0, bestc = sums[0];
#pragma unroll
        for (int d = 1; d < NDIRS; ++d)
            if (sums[d] > bestc) { bestc = sums[d]; best = d; }
        *out_last = (float)best;
    }
}

extern "C" void kernel_launch(void* const* d_in, const int* in_sizes, int n_in,
                              void* d_out, int out_size, void* d_ws, size_t ws_size,
                              hipStream_t stream) {
    const float* x   = (const float*)d_in[0];
    const int*   wid = (const int*)d_in[1];
    float*       out = (float*)d_out;
    int*         ws  = (int*)d_ws;

    wads_region_hist<<<NREG, 256, 0, stream>>>(wid, ws);

    constexpr int TOTAL4 = B * PL * LAT * LON * C4;   // 22,020,096 (multiple of 256)
    wads_gather<<<TOTAL4 / 256, 256, 0, stream>>>(x, ws, out);

    wads_finalize<<<1, 32, 0, stream>>>(ws, out + (out_size - 1));
}